// DeepState_44126493999562
// MI455X (gfx1250) — compile-verified
//
#include <hip/hip_runtime.h>
#include <hip/hip_bf16.h>

// ---------------------------------------------------------------------------
// DeepState (LSTM -> Kalman filter -> RTS smoother) for gfx1250 / MI455X.
// All matmuls (big GEMMs and the per-step 32x32x32 products in the scans)
// use V_WMMA_F32_16X16X4_F32 (fp32 kept throughout: the recursion contains
// a matrix inverse). Scans parallelize over the 128 independent batches,
// one wave32 per batch; per-step parameters are software-pipelined with
// float4 register prefetch to hide HBM latency at 1-wave occupancy.
// ---------------------------------------------------------------------------

#define Bb   128
#define Tt   128
#define DXx  64
#define Uu   128
#define DZz  32
#define G4U  512   // 4*U

typedef float v2f __attribute__((ext_vector_type(2)));
typedef float v8f __attribute__((ext_vector_type(8)));

#define SLD 36      // LDS row stride for 32x32 mats: 16B-aligned, bank-spread

__device__ __forceinline__ float sigmoidf(float x) {
    return 1.0f / (1.0f + __expf(-x));
}
__device__ __forceinline__ float tanh_fast(float x) {
    return 2.0f / (1.0f + __expf(-2.0f * x)) - 1.0f;
}

// ---------------------------------------------------------------------------
// Wave-level 32x32 matmul in LDS:  D = op(X) * op(Y) (+ Addm)
// X,Y,Addm,D: 32x32, row stride SLD, in LDS.
// ---------------------------------------------------------------------------
template <bool TA, bool TB, bool ADD>
__device__ __forceinline__ void mm32(const float* X, const float* Y,
                                     const float* Addm, float* D) {
    const int lane = threadIdx.x & 31;
    const int half = lane >> 4;
    const int l16  = lane & 15;
#pragma unroll
    for (int ti = 0; ti < 2; ++ti) {
#pragma unroll
        for (int tj = 0; tj < 2; ++tj) {
            const int n = tj * 16 + l16;
            v8f acc;
#pragma unroll
            for (int r = 0; r < 8; ++r)
                acc[r] = ADD ? Addm[(ti * 16 + r + 8 * half) * SLD + n] : 0.0f;
#pragma unroll
            for (int k0 = 0; k0 < 32; k0 += 4) {
                const int ka = k0 + 2 * half;
                v2f a, bb;
                if (TA) {
                    a.x = X[ka * SLD + ti * 16 + l16];
                    a.y = X[(ka + 1) * SLD + ti * 16 + l16];
                } else {
                    a.x = X[(ti * 16 + l16) * SLD + ka];
                    a.y = X[(ti * 16 + l16) * SLD + ka + 1];
                }
                if (TB) {
                    bb.x = Y[n * SLD + ka];
                    bb.y = Y[n * SLD + ka + 1];
                } else {
                    bb.x = Y[ka * SLD + n];
                    bb.y = Y[(ka + 1) * SLD + n];
                }
                acc = __builtin_amdgcn_wmma_f32_16x16x4_f32(
                    false, a, false, bb, (short)0, acc, false, false);
            }
#pragma unroll
            for (int r = 0; r < 8; ++r)
                D[(ti * 16 + r + 8 * half) * SLD + n] = acc[r];
        }
    }
}

// ---------------------------------------------------------------------------
// Generic row-major GEMM + bias: C[M,N] = A[M,K] @ B[K,N] + bias[N]
// One wave per 16x16 tile, block = (32,4).
// ---------------------------------------------------------------------------
__global__ __launch_bounds__(128) void wmma_gemm_bias_kernel(
    const float* __restrict__ A, const float* __restrict__ B,
    const float* __restrict__ bias, float* __restrict__ C,
    int M, int N, int K) {
    const int lane = threadIdx.x & 31;
    const int half = lane >> 4;
    const int l16  = lane & 15;
    const int mt = blockIdx.x;
    const int nt = blockIdx.y * blockDim.y + threadIdx.y;
    if (nt * 16 >= N) return;
    const int m0 = mt * 16, n0 = nt * 16;

    v8f acc;
    const float bv = bias ? bias[n0 + l16] : 0.0f;
#pragma unroll
    for (int r = 0; r < 8; ++r) acc[r] = bv;

    for (int k0 = 0; k0 < K; k0 += 4) {
        const int ka = k0 + 2 * half;
        v2f a, bb;
        a.x  = A[(size_t)(m0 + l16) * K + ka];
        a.y  = A[(size_t)(m0 + l16) * K + ka + 1];
        bb.x = B[(size_t)ka * N + n0 + l16];
        bb.y = B[(size_t)(ka + 1) * N + n0 + l16];
        acc = __builtin_amdgcn_wmma_f32_16x16x4_f32(
            false, a, false, bb, (short)0, acc, false, false);
    }
#pragma unroll
    for (int r = 0; r < 8; ++r)
        C[(size_t)(m0 + r + 8 * half) * N + n0 + l16] = acc[r];
}

// ---------------------------------------------------------------------------
// R projection (N = 1): per-row dot product.
// ---------------------------------------------------------------------------
__global__ void proj_R_kernel(const float* __restrict__ hs,
                              const float* __restrict__ WR,
                              const float* __restrict__ bR,
                              float* __restrict__ Rout, int rows) {
    int i = blockIdx.x * blockDim.x + threadIdx.x;
    if (i >= rows) return;
    float s = bR[0];
    for (int k = 0; k < Uu; ++k) s += hs[(size_t)i * Uu + k] * WR[k];
    Rout[i] = s;
}

// ---------------------------------------------------------------------------
// LSTM: 8 blocks x 256 threads (8 waves). Block owns 16 batches; h in LDS,
// c in registers; wave w owns gate-column slice [16w,16w+16) so i,f,g,o
// accumulators for its columns live in the same wave. xz gate biases for
// step t+1 are register-prefetched during step t's GEMM. Wh streams from
// global (off the WMMA dependency chain; pipelines from L2).
// ---------------------------------------------------------------------------
#define HLD 132

__global__ __launch_bounds__(256) void lstm_kernel(
    const float* __restrict__ xz,   // (B,T,4U) rows b*T+t
    const float* __restrict__ Wh,   // (U,4U)
    float* __restrict__ hs) {       // (T,B,U)
    __shared__ float hbuf[16 * HLD];
    const int lane = threadIdx.x & 31;
    const int w    = threadIdx.x >> 5;
    const int half = lane >> 4;
    const int l16  = lane & 15;
    const int bgrp = blockIdx.x;

    for (int i = threadIdx.x; i < 16 * HLD; i += 256) hbuf[i] = 0.0f;
    float c8[8];
#pragma unroll
    for (int r = 0; r < 8; ++r) c8[r] = 0.0f;

    float xr[4][8];
    {   // preload xz for t = 0
#pragma unroll
        for (int g = 0; g < 4; ++g)
#pragma unroll
            for (int r = 0; r < 8; ++r) {
                const int gb = bgrp * 16 + r + 8 * half;
                xr[g][r] = xz[((size_t)gb * Tt + 0) * G4U + g * Uu + w * 16 + l16];
            }
    }
    __syncthreads();

    for (int t = 0; t < Tt; ++t) {
        v8f z[4];
#pragma unroll
        for (int g = 0; g < 4; ++g)
#pragma unroll
            for (int r = 0; r < 8; ++r) z[g][r] = xr[g][r];

        if (t + 1 < Tt) {   // prefetch next step's gate biases (overlaps GEMM)
#pragma unroll
            for (int g = 0; g < 4; ++g)
#pragma unroll
                for (int r = 0; r < 8; ++r) {
                    const int gb = bgrp * 16 + r + 8 * half;
                    xr[g][r] = xz[((size_t)gb * Tt + (t + 1)) * G4U + g * Uu + w * 16 + l16];
                }
        }

#pragma unroll
        for (int g = 0; g < 4; ++g) {
            const int n0 = g * Uu + w * 16;
            for (int k0 = 0; k0 < Uu; k0 += 4) {
                const int ka = k0 + 2 * half;
                v2f a, bb;
                a.x  = hbuf[l16 * HLD + ka];
                a.y  = hbuf[l16 * HLD + ka + 1];
                bb.x = Wh[(size_t)ka * G4U + n0 + l16];
                bb.y = Wh[(size_t)(ka + 1) * G4U + n0 + l16];
                z[g] = __builtin_amdgcn_wmma_f32_16x16x4_f32(
                    false, a, false, bb, (short)0, z[g], false, false);
            }
        }
        __syncthreads();   // all reads of old h complete
        const int j = w * 16 + l16;
#pragma unroll
        for (int r = 0; r < 8; ++r) {
            const float iv = sigmoidf(z[0][r]);
            const float fv = sigmoidf(z[1][r]);
            const float gv = tanh_fast(z[2][r]);
            const float ov = sigmoidf(z[3][r]);
            const float c  = fv * c8[r] + iv * gv;
            c8[r] = c;
            const float h = ov * tanh_fast(c);
            const int m = r + 8 * half;
            hbuf[m * HLD + j] = h;
            const int gb = bgrp * 16 + m;
            hs[((size_t)t * Bb + gb) * Uu + j] = h;
        }
        __syncthreads();   // new h visible
    }
}

// ---------------------------------------------------------------------------
// Kalman forward filter: one wave per batch, sequential over T.
// A_t/Q_t/c_t/R_t/y_t for step t+1 are float4-prefetched into registers
// while step t computes (latency off the sequential chain).
// ---------------------------------------------------------------------------
__global__ __launch_bounds__(32) void kalman_forward_kernel(
    const float* __restrict__ Aall, const float* __restrict__ Call,
    const float* __restrict__ Qall, const float* __restrict__ Rall,
    const float* __restrict__ y,    const float* __restrict__ mu0,
    const float* __restrict__ sig0,
    float* __restrict__ mu_pred, float* __restrict__ Sig_pred,
    float* __restrict__ mu_filt, float* __restrict__ Sig_filt) {
    __shared__ float Sp[32 * SLD], Sf[32 * SLD], Am[32 * SLD], Qm[32 * SLD];
    __shared__ float T1[32 * SLD], T2m[32 * SLD], IK[32 * SLD];
    __shared__ float mup[32], muf[32], Kv[32], cv[32], vtmp[32];
    const int lane = threadIdx.x & 31;
    const int b = blockIdx.x;
    const float4* A4 = (const float4*)Aall;
    const float4* Q4 = (const float4*)Qall;

    for (int j = 0; j < 32; ++j)
        Sp[lane * SLD + j] = sig0[(size_t)b * 1024 + lane * 32 + j];
    mup[lane] = mu0[b * 32 + lane];

    float4 Ar[8], Qr[8];
    float cR, Rr, ytR;
    {   // preload t = 0
        const size_t base = (size_t)b;
#pragma unroll
        for (int i = 0; i < 8; ++i) {
            Ar[i] = A4[base * 256 + lane * 8 + i];
            Qr[i] = Q4[base * 256 + lane * 8 + i];
        }
        cR = Call[base * 32 + lane];
        Rr = Rall[base];
        ytR = y[(size_t)b * Tt + 0];
    }
    __syncthreads();

    for (int t = 0; t < Tt; ++t) {
        const size_t base = (size_t)t * Bb + b;
        // commit prefetched params to LDS
#pragma unroll
        for (int i = 0; i < 8; ++i) {
            *(float4*)&Am[lane * SLD + 4 * i] = Ar[i];
            *(float4*)&Qm[lane * SLD + 4 * i] = Qr[i];
        }
        cv[lane] = cR;
        const float R  = Rr;
        const float yt = ytR;
        __syncthreads();

        if (t + 1 < Tt) {   // prefetch next step's params
            const size_t b1 = (size_t)(t + 1) * Bb + b;
#pragma unroll
            for (int i = 0; i < 8; ++i) {
                Ar[i] = A4[b1 * 256 + lane * 8 + i];
                Qr[i] = Q4[b1 * 256 + lane * 8 + i];
            }
            cR = Call[b1 * 32 + lane];
            Rr = Rall[b1];
            ytR = y[(size_t)b * Tt + t + 1];
        }

        // v = Sig_p @ c ; s = c.v + R ; y_hat = c.mu_p
        float v = 0.0f;
        for (int k = 0; k < 32; ++k) v += Sp[lane * SLD + k] * cv[k];
        vtmp[lane] = v;
        __syncthreads();
        float s = R, yh = 0.0f;
        for (int k = 0; k < 32; ++k) { s += cv[k] * vtmp[k]; yh += cv[k] * mup[k]; }
        const float r  = yt - yh;
        const float Kl = vtmp[lane] / s;
        Kv[lane]  = Kl;
        const float mfl = mup[lane] + Kl * r;
        muf[lane] = mfl;
        __syncthreads();

        // IKC = I - K c^T
        for (int j = 0; j < 32; ++j)
            IK[lane * SLD + j] = (lane == j ? 1.0f : 0.0f) - Kl * cv[j];
        __syncthreads();

        mm32<false, false, false>(IK, Sp, nullptr, T1);   // T1 = IKC @ Sp
        __syncthreads();
        mm32<false, true, false>(T1, IK, nullptr, Sf);    // Sf = T1 @ IKC^T
        __syncthreads();
        for (int j = 0; j < 32; ++j) Sf[lane * SLD + j] += Kl * Kv[j] * R;
        __syncthreads();

        // store filtered (b128 stores)
#pragma unroll
        for (int i = 0; i < 8; ++i)
            ((float4*)&Sig_filt[base * 1024 + lane * 32])[i] =
                *(const float4*)&Sf[lane * SLD + 4 * i];
        mu_filt[base * 32 + lane] = mfl;

        // predict
        float mp = 0.0f;
        for (int k = 0; k < 32; ++k) mp += Am[lane * SLD + k] * muf[k];
        mm32<false, false, false>(Am, Sf, nullptr, T2m);  // T2 = A @ Sf
        __syncthreads();
        mm32<false, true, true>(T2m, Am, Qm, Sp);         // Sp = T2 @ A^T + Q
        __syncthreads();
        mup[lane] = mp;
        __syncthreads();
        mu_pred[base * 32 + lane] = mp;
#pragma unroll
        for (int i = 0; i < 8; ++i)
            ((float4*)&Sig_pred[base * 1024 + lane * 32])[i] =
                *(const float4*)&Sp[lane * SLD + 4 * i];
        __syncthreads();
    }
}

// ---------------------------------------------------------------------------
// RTS smoother (reverse scan): one wave per batch. 32x32 inverse via
// wave-parallel Gauss-Jordan in LDS; matmuls via WMMA. Sig_pred/Sig_filt/A
// for step t-1 register-prefetched during step t.
// ---------------------------------------------------------------------------
__global__ __launch_bounds__(32) void kalman_backward_kernel(
    const float* __restrict__ Aall,
    const float* __restrict__ mu_pred, const float* __restrict__ Sig_pred,
    const float* __restrict__ mu_filt, const float* __restrict__ Sig_filt,
    float* __restrict__ mu_out, float* __restrict__ Sig_out) {
    __shared__ float Sp[32 * SLD], Sf[32 * SLD], Am[32 * SLD], Inv[32 * SLD];
    __shared__ float Jm[32 * SLD], T1[32 * SLD], T2m[32 * SLD], Dm[32 * SLD];
    __shared__ float Sb[32 * SLD];
    __shared__ float Aug[32 * 65];
    __shared__ float mub[32], dv[32];
    const int lane = threadIdx.x & 31;
    const int b = blockIdx.x;
    const float4* A4  = (const float4*)Aall;
    const float4* SP4 = (const float4*)Sig_pred;
    const float4* SF4 = (const float4*)Sig_filt;

    const size_t lastb = (size_t)(Tt - 1) * Bb + b;
    for (int j = 0; j < 32; ++j)
        Sb[lane * SLD + j] = Sig_filt[lastb * 1024 + lane * 32 + j];
    mub[lane] = mu_filt[lastb * 32 + lane];
    mu_out[lastb * 32 + lane] = mub[lane];
#pragma unroll
    for (int i = 0; i < 8; ++i)
        ((float4*)&Sig_out[lastb * 1024 + lane * 32])[i] =
            *(const float4*)&Sb[lane * SLD + 4 * i];

    float4 Spr[8], Sfr[8], Amr[8];
    float mplR, mflR;
    {   // preload t = T-2
        const size_t b0 = (size_t)(Tt - 2) * Bb + b;
#pragma unroll
        for (int i = 0; i < 8; ++i) {
            Spr[i] = SP4[b0 * 256 + lane * 8 + i];
            Sfr[i] = SF4[b0 * 256 + lane * 8 + i];
            Amr[i] = A4 [b0 * 256 + lane * 8 + i];
        }
        mplR = mu_pred[b0 * 32 + lane];
        mflR = mu_filt[b0 * 32 + lane];
    }
    __syncthreads();

    for (int t = Tt - 2; t >= 0; --t) {
        const size_t base = (size_t)t * Bb + b;
        // commit prefetched params to LDS
#pragma unroll
        for (int i = 0; i < 8; ++i) {
            *(float4*)&Sp[lane * SLD + 4 * i] = Spr[i];
            *(float4*)&Sf[lane * SLD + 4 * i] = Sfr[i];
            *(float4*)&Am[lane * SLD + 4 * i] = Amr[i];
        }
        const float mpl = mplR;
        const float mfl = mflR;
        __syncthreads();

        if (t > 0) {   // prefetch next (t-1) step's params
            const size_t b1 = (size_t)(t - 1) * Bb + b;
#pragma unroll
            for (int i = 0; i < 8; ++i) {
                Spr[i] = SP4[b1 * 256 + lane * 8 + i];
                Sfr[i] = SF4[b1 * 256 + lane * 8 + i];
                Amr[i] = A4 [b1 * 256 + lane * 8 + i];
            }
            mplR = mu_pred[b1 * 32 + lane];
            mflR = mu_filt[b1 * 32 + lane];
        }

        // Gauss-Jordan: Inv = Sp^-1
        for (int j = 0; j < 64; ++j)
            Aug[lane * 65 + j] = (j < 32) ? Sp[lane * SLD + j]
                                          : ((j - 32) == lane ? 1.0f : 0.0f);
        __syncthreads();
        for (int k = 0; k < 32; ++k) {
            if (lane == k) {
                const float pinv = 1.0f / Aug[k * 65 + k];
                for (int j = k; j < 64; ++j) Aug[k * 65 + j] *= pinv;
            }
            __syncthreads();
            const float f = (lane == k) ? 0.0f : Aug[lane * 65 + k];
            for (int j = k; j < 64; ++j) Aug[lane * 65 + j] -= f * Aug[k * 65 + j];
            __syncthreads();
        }
        for (int j = 0; j < 32; ++j) Inv[lane * SLD + j] = Aug[lane * 65 + 32 + j];
        __syncthreads();

        mm32<true, false, false>(Am, Inv, nullptr, T1);   // T1 = A^T @ Inv
        __syncthreads();
        mm32<false, false, false>(Sf, T1, nullptr, Jm);   // J = Sf @ T1
        __syncthreads();

        dv[lane] = mub[lane] - mpl;
        for (int j = 0; j < 32; ++j)
            Dm[lane * SLD + j] = Sb[lane * SLD + j] - Sp[lane * SLD + j];
        __syncthreads();

        float mb = mfl;
        for (int k = 0; k < 32; ++k) mb += Jm[lane * SLD + k] * dv[k];

        mm32<false, false, false>(Jm, Dm, nullptr, T2m);  // T2 = J @ (Sb - Sp)
        __syncthreads();
        mm32<false, true, true>(T2m, Jm, Sf, Sb);         // Sb = Sf + T2 @ J^T
        __syncthreads();
        mub[lane] = mb;
        __syncthreads();

        mu_out[base * 32 + lane] = mb;
#pragma unroll
        for (int i = 0; i < 8; ++i)
            ((float4*)&Sig_out[base * 1024 + lane * 32])[i] =
                *(const float4*)&Sb[lane * SLD + 4 * i];
        __syncthreads();
    }
}

// ---------------------------------------------------------------------------
// Host launcher
// ---------------------------------------------------------------------------
extern "C" void kernel_launch(void* const* d_in, const int* in_sizes, int n_in,
                              void* d_out, int out_size, void* d_ws, size_t ws_size,
                              hipStream_t stream) {
    const float* x    = (const float*)d_in[0];
    const float* y    = (const float*)d_in[1];
    const float* Wx   = (const float*)d_in[2];
    const float* Wh   = (const float*)d_in[3];
    const float* bv   = (const float*)d_in[4];
    const float* WA   = (const float*)d_in[5];
    const float* bA   = (const float*)d_in[6];
    const float* WC   = (const float*)d_in[7];
    const float* bC   = (const float*)d_in[8];
    const float* WQ   = (const float*)d_in[9];
    const float* bQ   = (const float*)d_in[10];
    const float* WR   = (const float*)d_in[11];
    const float* bR   = (const float*)d_in[12];
    const float* Wmu  = (const float*)d_in[13];
    const float* bmu  = (const float*)d_in[14];
    const float* Wsig = (const float*)d_in[15];
    const float* bsig = (const float*)d_in[16];

    const size_t BT = (size_t)Bb * Tt;   // 16384
    float* ws = (float*)d_ws;
    size_t o = 0;
    float* xz    = ws + o; o += BT * G4U;
    float* hs    = ws + o; o += BT * Uu;
    float* Ap    = ws + o; o += BT * 1024;
    float* Cp    = ws + o; o += BT * 32;
    float* Qp    = ws + o; o += BT * 1024;
    float* Rp    = ws + o; o += BT;
    float* m0    = ws + o; o += (size_t)Bb * 32;
    float* s0    = ws + o; o += (size_t)Bb * 1024;
    float* mupr  = ws + o; o += BT * 32;
    float* mufl  = ws + o; o += BT * 32;
    float* Spred = ws + o; o += BT * 1024;
    float* Sfilt = ws + o; o += BT * 1024;

    float* mu_out  = (float*)d_out;
    float* Sig_out = mu_out + BT * 32;

    dim3 blk(32, 4, 1);

    // 1) xz = x @ Wx + b
    wmma_gemm_bias_kernel<<<dim3(BT / 16, 8), blk, 0, stream>>>(
        x, Wx, bv, xz, (int)BT, G4U, DXx);

    // 2) LSTM recurrence -> hs (T,B,U)
    lstm_kernel<<<dim3(Bb / 16), dim3(256), 0, stream>>>(xz, Wh, hs);

    // 3) Projections
    wmma_gemm_bias_kernel<<<dim3(BT / 16, 16), blk, 0, stream>>>(
        hs, WA, bA, Ap, (int)BT, 1024, Uu);
    wmma_gemm_bias_kernel<<<dim3(BT / 16, 1), blk, 0, stream>>>(
        hs, WC, bC, Cp, (int)BT, 32, Uu);
    wmma_gemm_bias_kernel<<<dim3(BT / 16, 16), blk, 0, stream>>>(
        hs, WQ, bQ, Qp, (int)BT, 1024, Uu);
    proj_R_kernel<<<dim3((BT + 255) / 256), dim3(256), 0, stream>>>(
        hs, WR, bR, Rp, (int)BT);
    const float* out1 = hs + (size_t)1 * Bb * Uu;   // h[:,1]
    wmma_gemm_bias_kernel<<<dim3(Bb / 16, 1), blk, 0, stream>>>(
        out1, Wmu, bmu, m0, Bb, 32, Uu);
    wmma_gemm_bias_kernel<<<dim3(Bb / 16, 16), blk, 0, stream>>>(
        out1, Wsig, bsig, s0, Bb, 1024, Uu);

    // 4) Forward Kalman filter
    kalman_forward_kernel<<<dim3(Bb), dim3(32), 0, stream>>>(
        Ap, Cp, Qp, Rp, y, m0, s0, mupr, Spred, mufl, Sfilt);

    // 5) RTS smoother -> outputs
    kalman_backward_kernel<<<dim3(Bb), dim3(32), 0, stream>>>(
        Ap, mupr, Spred, mufl, Sfilt, mu_out, Sig_out);
}